// ConvDGN_16286515986845
// MI455X (gfx1250) — compile-verified
//
#include <hip/hip_runtime.h>

#define IN_DIM    128
#define OUT_DIM   128
#define NUM_REL   20
#define NUM_BASES 10
#define KZ   (NUM_BASES * IN_DIM)      /* 1280: z part of the K dimension      */
#define KTOT (KZ + IN_DIM)             /* 1408: [z | x] concatenated K         */
#define LDS_STRIDE (KTOT + 4)          /* 1412 dwords: conflict-free A rows    */

typedef __attribute__((ext_vector_type(2))) float v2f;
typedef __attribute__((ext_vector_type(8))) float v8f;

// ---------------------------------------------------------------------------
// 1) zero the workspace (cnt + sums), grid-stride
// ---------------------------------------------------------------------------
__global__ void rgcn_zero_kernel(unsigned* __restrict__ p, size_t nwords) {
    size_t i      = (size_t)blockIdx.x * blockDim.x + threadIdx.x;
    size_t stride = (size_t)gridDim.x * blockDim.x;
    for (; i < nwords; i += stride) p[i] = 0u;
}

// ---------------------------------------------------------------------------
// 2) cnt[dst*R + rel] += 1 per edge
// ---------------------------------------------------------------------------
__global__ void rgcn_count_kernel(const int* __restrict__ dst,
                                  const int* __restrict__ rel,
                                  unsigned* __restrict__ cnt, int E) {
    int e = blockIdx.x * blockDim.x + threadIdx.x;
    if (e < E) atomicAdd(&cnt[dst[e] * NUM_REL + rel[e]], 1u);
}

// ---------------------------------------------------------------------------
// 3) edge scatter: one wave32 per edge, raw neighbor-feature sums.
//    S[dst, rel, :] += x[src, :]   (4 f32 atomics per lane = 128 per edge)
// ---------------------------------------------------------------------------
__global__ __launch_bounds__(256)
void rgcn_scatter_kernel(const float* __restrict__ x,
                         const int* __restrict__ src,
                         const int* __restrict__ dst,
                         const int* __restrict__ rel,
                         float* __restrict__ sums, int E) {
    int e    = (int)((blockIdx.x * blockDim.x + threadIdx.x) >> 5);
    int lane = threadIdx.x & 31;
    if (e >= E) return;
    int s = src[e], d = dst[e], r = rel[e];
    const float4 xv = *(const float4*)(x + (size_t)s * IN_DIM + lane * 4);
    float* p = sums + ((size_t)d * NUM_REL + r) * IN_DIM + lane * 4;
    unsafeAtomicAdd(p + 0, xv.x);
    unsafeAtomicAdd(p + 1, xv.y);
    unsafeAtomicAdd(p + 2, xv.z);
    unsafeAtomicAdd(p + 3, xv.w);
}

// ---------------------------------------------------------------------------
// 4) dense phase per 16-node tile:
//    a) stage invcnt[16][20] and comp[20][10] in LDS,
//    b) build A tile rows [ z[i,0,:] .. z[i,9,:] | x[i,:] ] where
//       z[i,b,:] = sum_r comp[r,b]*invcnt[i,r]*S[i,r,:]   (folded here),
//    c) out = A @ [basis;root] + bias via chained v_wmma_f32_16x16x4_f32.
// ---------------------------------------------------------------------------
__global__ __launch_bounds__(256)
void rgcn_gemm_kernel(const float* __restrict__ sums,
                      const float* __restrict__ x,
                      const unsigned* __restrict__ cnt,
                      const float* __restrict__ comp,
                      const float* __restrict__ basis,
                      const float* __restrict__ root,
                      const float* __restrict__ bias,
                      float* __restrict__ out, int N) {
    extern __shared__ float lds[];
    float* At   = lds;                         // 16 * LDS_STRIDE floats
    float* invc = At + 16 * LDS_STRIDE;        // 16 * NUM_REL
    float* scmp = invc + 16 * NUM_REL;         // NUM_REL * NUM_BASES
    const int node0 = blockIdx.x * 16;
    const int tid   = threadIdx.x;

    // (a) small per-block tables
    if (tid < 16 * NUM_REL) {
        int i = tid / NUM_REL, r = tid % NUM_REL;
        int node = node0 + i; if (node >= N) node = N - 1;
        unsigned c = cnt[(size_t)node * NUM_REL + r];
        invc[tid] = 1.0f / (float)(c ? c : 1u);
    }
    if (tid < NUM_REL * NUM_BASES) scmp[tid] = comp[tid];
    __syncthreads();

    // (b) build A tile: 16 rows x 352 float4 granules
    const int G = KTOT / 4;                    // 352 (320 z + 32 x)
    for (int g = tid; g < 16 * G; g += 256) {
        int row = g / G, c4 = g % G;
        int node = node0 + row; if (node >= N) node = N - 1;
        if (c4 < KZ / 4) {                     // z part: contract S over relations
            int b  = c4 / (IN_DIM / 4);
            int kk = (c4 % (IN_DIM / 4)) * 4;
            const float* srow = sums + ((size_t)node * NUM_REL) * IN_DIM + kk;
            float4 acc = {0.f, 0.f, 0.f, 0.f};
#pragma unroll
            for (int r = 0; r < NUM_REL; ++r) {
                float w = scmp[r * NUM_BASES + b] * invc[row * NUM_REL + r];
                float4 sv = *(const float4*)(srow + (size_t)r * IN_DIM);
                acc.x += w * sv.x; acc.y += w * sv.y;
                acc.z += w * sv.z; acc.w += w * sv.w;
            }
            *(float4*)(At + row * LDS_STRIDE + c4 * 4) = acc;
        } else {                               // x part: straight copy
            int k = (c4 - KZ / 4) * 4;
            *(float4*)(At + row * LDS_STRIDE + KZ + k) =
                *(const float4*)(x + (size_t)node * IN_DIM + k);
        }
    }
    __syncthreads();

    // (c) WMMA main loops
    const int wave = tid >> 5, lane = tid & 31;
    const int n0 = wave * 16 + (lane & 15);    // output column
    const int m  = lane & 15;                  // A row within tile
    const int kh = (lane >> 4) * 2;            // K sub-offset per ISA layout

    v8f c = {0.f,0.f,0.f,0.f,0.f,0.f,0.f,0.f};
    const float* Arow = At + m * LDS_STRIDE + kh;
    const float* bp   = basis + (size_t)kh * OUT_DIM + n0;
#pragma unroll 4
    for (int kk = 0; kk < KZ; kk += 4) {       // z @ basis (K = 1280)
        v2f a; a.x = Arow[0]; a.y = Arow[1];
        v2f b; b.x = bp[0];   b.y = bp[OUT_DIM];
        c = __builtin_amdgcn_wmma_f32_16x16x4_f32(false, a, false, b,
                                                  (short)0, c, false, false);
        __builtin_prefetch(bp + 16 * OUT_DIM, 0, 1);   // global_prefetch_b8
        Arow += 4; bp += 4 * OUT_DIM;
    }
    const float* rp = root + (size_t)kh * OUT_DIM + n0;
#pragma unroll 4
    for (int kk = 0; kk < IN_DIM; kk += 4) {   // x @ root (K = 128)
        v2f a; a.x = Arow[0]; a.y = Arow[1];
        v2f b; b.x = rp[0];   b.y = rp[OUT_DIM];
        c = __builtin_amdgcn_wmma_f32_16x16x4_f32(false, a, false, b,
                                                  (short)0, c, false, false);
        Arow += 4; rp += 4 * OUT_DIM;
    }

    // C/D layout: VGPR i -> M=i (lanes 0-15) / M=8+i (lanes 16-31), N=lane%16.
    const float bv = bias[n0];
    const int rbase = (lane >> 4) * 8;
#pragma unroll
    for (int i = 0; i < 8; ++i) {
        int row = node0 + rbase + i;
        if (row < N) out[(size_t)row * OUT_DIM + n0] = c[i] + bv;
    }
}

// ---------------------------------------------------------------------------
extern "C" void kernel_launch(void* const* d_in, const int* in_sizes, int n_in,
                              void* d_out, int out_size, void* d_ws, size_t ws_size,
                              hipStream_t stream) {
    const float* x     = (const float*)d_in[0];
    const int*   ei    = (const int*)  d_in[1];
    const int*   et    = (const int*)  d_in[2];
    const float* basis = (const float*)d_in[3];
    const float* comp  = (const float*)d_in[4];
    const float* root  = (const float*)d_in[5];
    const float* bias  = (const float*)d_in[6];
    float*       out   = (float*)d_out;

    const int N = in_sizes[0] / IN_DIM;
    const int E = in_sizes[1] / 2;
    const int* src = ei;
    const int* dst = ei + E;

    // Workspace layout: [cnt: N*R u32 | S: N*R*IN f32]
    unsigned* cnt = (unsigned*)d_ws;
    size_t cnt_bytes = (((size_t)N * NUM_REL * sizeof(unsigned)) + 255) & ~(size_t)255;
    float* sums = (float*)((char*)d_ws + cnt_bytes);
    size_t zero_words = cnt_bytes / 4 + (size_t)N * NUM_REL * IN_DIM;

    rgcn_zero_kernel<<<4096, 256, 0, stream>>>((unsigned*)d_ws, zero_words);
    rgcn_count_kernel<<<(E + 255) / 256, 256, 0, stream>>>(dst, et, cnt, E);
    rgcn_scatter_kernel<<<(E + 7) / 8, 256, 0, stream>>>(x, src, dst, et, sums, E);

    size_t lds_bytes = (size_t)(16 * LDS_STRIDE + 16 * NUM_REL +
                                NUM_REL * NUM_BASES) * sizeof(float);  // 92448 B
    rgcn_gemm_kernel<<<(N + 15) / 16, 256, lds_bytes, stream>>>(
        sums, x, cnt, comp, basis, root, bias, out, N);
}